// Net_90357521973328
// MI455X (gfx1250) — compile-verified
//
#include <hip/hip_runtime.h>
#include <math.h>

// MI455X / gfx1250, wave32. Grouped per-object 3-layer MLP (GEMV chain).
// HBM-bound (~0.52 FLOP/byte, ~770 MB streamed once -> ~33us floor @ 23.3 TB/s).
// Weights are staged global->LDS with CDNA5 async-to-LDS (perfectly coalesced
// 128B-line reads, double buffered, s_wait_asynccnt), then fed to
// v_wmma_f32_16x16x4_f32 from a bank-conflict-free padded LDS tile.

typedef __attribute__((ext_vector_type(2))) float v2f;
typedef __attribute__((ext_vector_type(8))) float v8f;

#define N_OBJ   2048
#define IN_DIM  128
#define MID_DIM 256

#define KC   32                 // K-chunk per staging step (32 floats = 128B/row)
#define SP   44                 // padded LDS row stride (floats): 176B = 16B-aligned,
                                // and 44*n mod 64 distinct for n=0..15 -> no bank conflicts
#define RW   32                 // rows owned per wave (2 WMMA tiles)
#define BUF_FLOATS (RW * SP)    // 1408 floats per staging buffer
#define BUF_BYTES  (BUF_FLOATS * 4)

__device__ __forceinline__ float sigmoidf(float t) {
    return 1.0f / (1.0f + expf(-t));
}

__device__ __forceinline__ v8f wmma4(v2f a, v2f b, v8f c) {
    return __builtin_amdgcn_wmma_f32_16x16x4_f32(false, a, false, b,
                                                 (short)0, c, false, false);
}

// Async DMA, per lane: copies 16 bytes from the global address gptr to LDS byte offset lds_byte_off.
__device__ __forceinline__ void async_copy_b128(uint32_t lds_byte_off, const float* gptr) {
    asm volatile("global_load_async_to_lds_b128 %0, %1, off"
                 :: "v"(lds_byte_off), "v"((unsigned long long)(uintptr_t)gptr)
                 : "memory");
}

// y[row0], y[row1] = (W_blk @ in)[n], rows striped 16-wide across lanes.
// W_blk points at this wave's 32-row block; in_lds holds the K-float input.
template<int K>
__device__ __forceinline__ void matvec2(const float* __restrict__ Wblk,
                                        const float* in_lds,
                                        float* wbuf,          // 2*BUF_FLOATS in LDS
                                        int lane, float& r0, float& r1)
{
    const int srow = lane >> 3;          // staging: 4 rows / instruction, 8 lanes each
    const int scol = (lane & 7) << 2;    // staging: 16B (4 floats) per lane
    const int h    = lane >> 4;          // WMMA half-wave -> K pair select
    const int n    = lane & 15;          // WMMA N index = row within tile
    const uint32_t lb0 = (uint32_t)(uintptr_t)wbuf;   // low 32 bits == LDS byte offset
    const uint32_t lb1 = lb0 + BUF_BYTES;
    constexpr int NCH = K / KC;

    // Prime the pipeline: stage chunk 0 into buffer 0.
#pragma unroll
    for (int i = 0; i < 8; ++i) {
        const int rl = 4 * i + srow;
        async_copy_b128(lb0 + (uint32_t)((rl * SP + scol) << 2),
                        Wblk + (size_t)rl * K + scol);
    }

    v8f c0 = {}; v8f c1 = {};
#pragma unroll
    for (int c = 0; c < NCH; ++c) {
        if (c + 1 < NCH) {
            const uint32_t lbn = ((c + 1) & 1) ? lb1 : lb0;
#pragma unroll
            for (int i = 0; i < 8; ++i) {
                const int rl = 4 * i + srow;
                async_copy_b128(lbn + (uint32_t)((rl * SP + scol) << 2),
                                Wblk + (size_t)rl * K + (c + 1) * KC + scol);
            }
            // Next chunk's 8 in flight; wait until current chunk's 8 completed.
            asm volatile("s_wait_asynccnt 0x8" ::: "memory");
        } else {
            asm volatile("s_wait_asynccnt 0x0" ::: "memory");
        }

        const float* wb  = wbuf + (c & 1) * BUF_FLOATS;
        const float* pa  = in_lds + c * KC + 2 * h;
        const float* pb0 = wb + n * SP + 2 * h;       // tile0: local rows 0..15
        const float* pb1 = pb0 + 16 * SP;             // tile1: local rows 16..31
#pragma unroll
        for (int kk = 0; kk < KC; kk += 4) {
            v2f a  = *(const v2f*)(pa  + kk);  // A[m][k'] = in[k+k'] (all m identical)
            v2f b0 = *(const v2f*)(pb0 + kk);  // B[k'][n] = W[row0+n][k+k']
            v2f b1 = *(const v2f*)(pb1 + kk);
            c0 = wmma4(a, b0, c0);
            c1 = wmma4(a, b1, c1);
        }
    }
    // All D rows identical; D[0][n] is acc VGPR0, lanes 0..15.
    r0 = c0[0];
    r1 = c1[0];
}

__global__ __launch_bounds__(256) void mlp_grouped_kernel(
    const float* __restrict__ x,
    const float* __restrict__ W1, const float* __restrict__ b1,
    const float* __restrict__ W2, const float* __restrict__ b2,
    const float* __restrict__ W3, const float* __restrict__ b3,
    float* __restrict__ out)
{
    __shared__ __align__(16) float wstage[8][2 * BUF_FLOATS]; // 90112 B staging
    __shared__ __align__(16) float x_lds[IN_DIM];
    __shared__ __align__(16) float y1_lds[MID_DIM];           // reused for reduction
    __shared__ __align__(16) float y2_lds[MID_DIM];

    const int o    = blockIdx.x;
    const int tid  = threadIdx.x;
    const int lane = tid & 31;
    const int wave = tid >> 5;
    const int n    = lane & 15;

    if (tid < IN_DIM) x_lds[tid] = x[(size_t)o * IN_DIM + tid];
    __syncthreads();

    const int row0 = wave * RW + n;
    const int row1 = row0 + 16;
    float* wbuf = &wstage[wave][0];

    // ------- Layer 1: y1 = W1[o] @ x + b1[o]   (K=128, NO activation) -------
    {
        const float* Wblk = W1 + (size_t)o * MID_DIM * IN_DIM + (size_t)(wave * RW) * IN_DIM;
        float r0, r1;
        matvec2<IN_DIM>(Wblk, x_lds, wbuf, lane, r0, r1);
        if (lane < 16) {
            y1_lds[row0] = r0 + b1[(size_t)o * MID_DIM + row0];
            y1_lds[row1] = r1 + b1[(size_t)o * MID_DIM + row1];
        }
    }
    __syncthreads();

    // ------- Layer 2: y2 = sigmoid(W2[o] @ y1 + b2[o])   (K=256) -------
    {
        const float* Wblk = W2 + (size_t)o * MID_DIM * MID_DIM + (size_t)(wave * RW) * MID_DIM;
        float r0, r1;
        matvec2<MID_DIM>(Wblk, y1_lds, wbuf, lane, r0, r1);
        __syncthreads();   // everyone done reading y1 before it becomes scratch
        if (lane < 16) {
            y2_lds[row0] = sigmoidf(r0 + b2[(size_t)o * MID_DIM + row0]);
            y2_lds[row1] = sigmoidf(r1 + b2[(size_t)o * MID_DIM + row1]);
        }
    }
    __syncthreads();

    // ------- Layer 3: out = sigmoid(dot(W3[o], y2) + b3[o]) -------
    float* red = y1_lds;   // reuse as reduction scratch
    red[tid] = W3[(size_t)o * MID_DIM + tid] * y2_lds[tid];
    __syncthreads();
#pragma unroll
    for (int s = 128; s > 32; s >>= 1) {
        if (tid < s) red[tid] += red[tid + s];
        __syncthreads();
    }
    if (tid < 32) {
        float v = red[tid] + red[tid + 32];
#pragma unroll
        for (int d = 16; d > 0; d >>= 1) v += __shfl_down(v, d, 32);
        if (tid == 0) out[o] = sigmoidf(v + b3[o]);
    }
}

extern "C" void kernel_launch(void* const* d_in, const int* in_sizes, int n_in,
                              void* d_out, int out_size, void* d_ws, size_t ws_size,
                              hipStream_t stream) {
    (void)in_sizes; (void)n_in; (void)d_ws; (void)ws_size; (void)out_size;
    const float* x  = (const float*)d_in[0];
    const float* W1 = (const float*)d_in[1];
    const float* b1 = (const float*)d_in[2];
    const float* W2 = (const float*)d_in[3];
    const float* b2 = (const float*)d_in[4];
    const float* W3 = (const float*)d_in[5];
    const float* b3 = (const float*)d_in[6];
    float* out = (float*)d_out;

    dim3 grid(N_OBJ), block(256);
    mlp_grouped_kernel<<<grid, block, 0, stream>>>(x, W1, b1, W2, b2, W3, b3, out);
}